// Attention_12206297055554
// MI455X (gfx1250) — compile-verified
//
#include <hip/hip_runtime.h>

// MI455X / gfx1250, wave32. fp32 end-to-end via V_WMMA_F32_16X16X4_F32.
// HBM-bound workload (~150 MB min traffic @ 23.3 TB/s); intermediates live in
// d_ws and fit in 192 MB L2. GEMMs use the CDNA5 async-to-LDS pipeline
// (GLOBAL_LOAD_ASYNC_TO_LDS_B128 + s_wait_asynccnt) with LDS double-buffering.

typedef __attribute__((ext_vector_type(2))) float v2f;
typedef __attribute__((ext_vector_type(8))) float v8f;

#define SEQ   2048
#define DIMM  2048
#define NHEAD 16
#define HDIM  128
#define BATCH 2

#define KB      16            // k-slice per LDS stage
#define LDS_LD  20            // padded LDS row stride (floats): bank-spread, 80B (16B-aligned)
#define TILE_F  (64 * LDS_LD) // floats per 64-row staged panel
#define BUF_F   (2 * TILE_F)  // A panel + B panel per buffer

__device__ __forceinline__ v8f wmma4(v2f a, v2f b, v8f c) {
  return __builtin_amdgcn_wmma_f32_16x16x4_f32(false, a, false, b, (short)0, c,
                                               false, false);
}

// CDNA5 async copy: global -> LDS, 16B per lane, tracked by ASYNCcnt.
__device__ __forceinline__ void asyncB128(unsigned ldsByte, const float* g) {
  asm volatile("global_load_async_to_lds_b128 %0, %1, off"
               :: "v"(ldsByte), "v"((unsigned long long)(uintptr_t)g)
               : "memory");
}
__device__ __forceinline__ void waitAsync0() {
  asm volatile("s_wait_asynccnt 0x0" ::: "memory");
}

// ---------------------------------------------------------------------------
// C[M,N] = A[M,K] * B[N,K]^T. Workgroup (4 waves) -> 64x64 tile; per wave
// 16(M) x 64(N). K staged through double-buffered LDS via async b128 copies.
// ---------------------------------------------------------------------------
__global__ void __launch_bounds__(128)
gemm_tn_wmma_async(const float* __restrict__ A, const float* __restrict__ Bm,
                   float* __restrict__ C, int M, int N, int K) {
  __shared__ float smem[2 * BUF_F];
  const int tid  = threadIdx.x;
  const int lane = tid & 31;
  const int l15  = lane & 15;
  const int hi   = lane >> 4;
  const int wv   = tid >> 5;
  const int nTiles = N >> 6;
  const int m0 = (int)(blockIdx.x / nTiles) << 6;
  const int n0 = (int)(blockIdx.x % nTiles) << 6;

  // staging map: thread -> (row = tid>>1, col8 = (tid&1)*8) of the 64x16 panel
  const int sr = tid >> 1;
  const int sc = (tid & 1) * 8;
  const float* gA = A  + (size_t)(m0 + sr) * K + sc;
  const float* gB = Bm + (size_t)(n0 + sr) * K + sc;
  const unsigned smemBase = (unsigned)(uintptr_t)&smem[0]; // low 32b == LDS addr
  const unsigned stA = smemBase + (unsigned)(sr * LDS_LD + sc) * 4u;
  const unsigned stB = stA + (unsigned)TILE_F * 4u;

  v8f accs[4] = {};

  // prologue: stage k-block 0 into buffer 0
  asyncB128(stA, gA);      asyncB128(stA + 16, gA + 4);
  asyncB128(stB, gB);      asyncB128(stB + 16, gB + 4);

  const int nK = K >> 4;
  for (int kb = 0; kb < nK; ++kb) {
    waitAsync0();        // this wave's stage(kb) writes have landed in LDS
    __syncthreads();     // all waves' stages landed; prev compute fully done
    if (kb + 1 < nK) {   // issue stage(kb+1) into the other buffer
      const unsigned bo = (unsigned)((kb + 1) & 1) * (unsigned)BUF_F * 4u;
      const int k0 = (kb + 1) << 4;
      asyncB128(stA + bo, gA + k0);  asyncB128(stA + bo + 16, gA + k0 + 4);
      asyncB128(stB + bo, gB + k0);  asyncB128(stB + bo + 16, gB + k0 + 4);
    }
    const float* bufA = &smem[(kb & 1) * BUF_F];
    const float* bufB = bufA + TILE_F;
#pragma unroll
    for (int c = 0; c < 4; ++c) {
      v2f a = *(const v2f*)(bufA + (wv * 16 + l15) * LDS_LD + 4 * c + 2 * hi);
#pragma unroll
      for (int t = 0; t < 4; ++t) {
        v2f b = *(const v2f*)(bufB + (t * 16 + l15) * LDS_LD + 4 * c + 2 * hi);
        accs[t] = wmma4(a, b, accs[t]);
      }
    }
  }

  // C/D layout: element (Mr,Nc): VGPR v = Mr&7, lane = Nc + 16*(Mr>>3)
#pragma unroll
  for (int t = 0; t < 4; ++t) {
#pragma unroll
    for (int v = 0; v < 8; ++v) {
      C[(size_t)(m0 + wv * 16 + v + 8 * hi) * N + (n0 + t * 16 + l15)] =
          accs[t][v];
    }
  }
  (void)M;
}

// ---------------------------------------------------------------------------
// Flash attention (causal). qkv layout [b*S, 6144] = [b,n,3,h,hd].
// One wave owns 16 queries x 128 for one (b,h); streams 16-key blocks with
// online softmax. Fragment loads batched/prefetched to amortize waits.
// ---------------------------------------------------------------------------
__global__ void __launch_bounds__(128)
flash_attn_wmma(const float* __restrict__ qkv, float* __restrict__ out) {
  __shared__ float pTile[4][16 * LDS_LD];          // per-wave P staging, padded
  const int lane = threadIdx.x & 31;
  const int l15  = lane & 15;
  const int hi   = lane >> 4;
  const int wv   = threadIdx.x >> 5;
  const int gw   = blockIdx.x * 4 + wv;
  const int qb = gw & 127;
  const int h  = (gw >> 7) & 15;
  const int b  = gw >> 11;

  const size_t rowStride = 3 * DIMM;
  const float* Qb = qkv + (size_t)b * SEQ * rowStride + (size_t)h * HDIM;
  const float* Kb = Qb + DIMM;
  const float* Vb = Qb + 2 * DIMM;

  v2f qf[32];
  {
    const float* qrow = Qb + (size_t)(qb * 16 + l15) * rowStride + 2 * hi;
#pragma unroll
    for (int c = 0; c < 32; ++c) qf[c] = *(const v2f*)(qrow + 4 * c);
  }

  v8f o[8] = {};
  float mi[8], li[8];
#pragma unroll
  for (int v = 0; v < 8; ++v) { mi[v] = -__builtin_inff(); li[v] = 0.0f; }

  const float scale = 0.08838834764831845f;        // 128^-0.5

  for (int kb = 0; kb <= qb; ++kb) {
    // ---- S = Q*K^T : K fragments loaded one group (8) ahead of use ----
    const float* krow = Kb + (size_t)(kb * 16 + l15) * rowStride + 2 * hi;
    v2f kf[32];
#pragma unroll
    for (int j = 0; j < 8; ++j) kf[j] = *(const v2f*)(krow + 4 * j);
    v8f s = {};
#pragma unroll
    for (int g = 0; g < 4; ++g) {
      if (g < 3) {
#pragma unroll
        for (int j = 0; j < 8; ++j)
          kf[8 * (g + 1) + j] = *(const v2f*)(krow + 4 * (8 * (g + 1) + j));
      }
#pragma unroll
      for (int j = 0; j < 8; ++j) s = wmma4(qf[8 * g + j], kf[8 * g + j], s);
    }

    // ---- online softmax; S element (q = v+8*hi, key = l15) ----
    const int key = kb * 16 + l15;
#pragma unroll
    for (int v = 0; v < 8; ++v) {
      const int q = qb * 16 + v + 8 * hi;
      float sv = s[v] * scale;
      if (key > q) sv = -__builtin_inff();
      float rmax = sv;
      rmax = fmaxf(rmax, __shfl_xor(rmax, 1, 32));
      rmax = fmaxf(rmax, __shfl_xor(rmax, 2, 32));
      rmax = fmaxf(rmax, __shfl_xor(rmax, 4, 32));
      rmax = fmaxf(rmax, __shfl_xor(rmax, 8, 32));
      float mNew = fmaxf(mi[v], rmax);
      float p    = __expf(sv - mNew);
      float corr = __expf(mi[v] - mNew);
      float rsum = p;
      rsum += __shfl_xor(rsum, 1, 32);
      rsum += __shfl_xor(rsum, 2, 32);
      rsum += __shfl_xor(rsum, 4, 32);
      rsum += __shfl_xor(rsum, 8, 32);
      li[v] = li[v] * corr + rsum;
      mi[v] = mNew;
#pragma unroll
      for (int nb = 0; nb < 8; ++nb) o[nb][v] *= corr;
      pTile[wv][(v + 8 * hi) * LDS_LD + l15] = p;
    }
    __builtin_amdgcn_wave_barrier();               // keep DS store->load order

    v2f pf[4];
    const float* pb = &pTile[wv][l15 * LDS_LD + 2 * hi];
#pragma unroll
    for (int c = 0; c < 4; ++c) pf[c] = *(const v2f*)(pb + 4 * c);

    // ---- O += P*V : V fragments batched 16-at-a-time before WMMA bursts ----
#pragma unroll
    for (int half8 = 0; half8 < 2; ++half8) {
      v2f vt[16];
#pragma unroll
      for (int nb = 0; nb < 4; ++nb)
#pragma unroll
        for (int c = 0; c < 4; ++c) {
          const float* vp = Vb +
              (size_t)(kb * 16 + 4 * c + 2 * hi) * rowStride +
              (half8 * 4 + nb) * 16 + l15;
          v2f vf; vf.x = vp[0]; vf.y = vp[rowStride];
          vt[nb * 4 + c] = vf;
        }
#pragma unroll
      for (int nb = 0; nb < 4; ++nb)
#pragma unroll
        for (int c = 0; c < 4; ++c)
          o[half8 * 4 + nb] = wmma4(pf[c], vt[nb * 4 + c], o[half8 * 4 + nb]);
    }
  }

  // ---- epilogue ----
#pragma unroll
  for (int v = 0; v < 8; ++v) {
    const float inv = 1.0f / li[v];
    const size_t row = (size_t)b * SEQ + (size_t)(qb * 16 + v + 8 * hi);
#pragma unroll
    for (int nb = 0; nb < 8; ++nb)
      out[row * DIMM + h * HDIM + nb * 16 + l15] = o[nb][v] * inv;
  }
}

// ---------------------------------------------------------------------------
extern "C" void kernel_launch(void* const* d_in, const int* in_sizes, int n_in,
                              void* d_out, int out_size, void* d_ws,
                              size_t ws_size, hipStream_t stream) {
  const float* x     = (const float*)d_in[0];   // [2,2048,2048]
  const float* w_qkv = (const float*)d_in[1];   // [6144,2048]
  const float* w_out = (const float*)d_in[2];   // [2048,2048]
  float* outp = (float*)d_out;                  // [2,2048,2048]

  float* qkv  = (float*)d_ws;                   // 4096*6144 f32 (100.7 MB)
  float* attn = qkv + (size_t)4096 * 6144;      // 4096*2048 f32 ( 33.6 MB)
  (void)in_sizes; (void)n_in; (void)out_size; (void)ws_size;

  // 1) QKV projection: [4096,2048] @ [6144,2048]^T -> [4096,6144]
  {
    const int M = BATCH * SEQ, N = 3 * DIMM, K = DIMM;
    const int blocks = (M / 64) * (N / 64);     // 6144
    gemm_tn_wmma_async<<<dim3(blocks), dim3(128), 0, stream>>>(x, w_qkv, qkv,
                                                               M, N, K);
  }
  // 2) Causal flash attention over qkv -> attn [4096, 2048]
  {
    const int waves = BATCH * NHEAD * (SEQ / 16);   // 4096
    flash_attn_wmma<<<dim3(waves / 4), dim3(128), 0, stream>>>(qkv, attn);
  }
  // 3) Output projection: [4096,2048] @ [2048,2048]^T -> d_out
  {
    const int M = BATCH * SEQ, N = DIMM, K = DIMM;
    const int blocks = (M / 64) * (N / 64);     // 2048
    gemm_tn_wmma_async<<<dim3(blocks), dim3(128), 0, stream>>>(attn, w_out,
                                                               outp, M, N, K);
  }
}